// QuanvolutionFilter_65481071410265
// MI455X (gfx1250) — compile-verified
//
#include <hip/hip_runtime.h>

typedef __attribute__((ext_vector_type(2))) float v2f;
typedef __attribute__((ext_vector_type(8))) float v8f;

// ---- tiny 16-dim circuit simulator (runs on 16 threads/block to build U) ----
__device__ __forceinline__ void apply_cnot16(float* s, int ctrl, int tgt) {
    const int bc = 3 - ctrl, bt = 3 - tgt;
#pragma unroll
    for (int m = 0; m < 16; ++m) {
        if (((m >> bc) & 1) && !((m >> bt) & 1)) {
            const int m1 = m | (1 << bt);
            float t = s[m]; s[m] = s[m1]; s[m1] = t;
        }
    }
}

__device__ __forceinline__ void apply_ry16(float* s, float th, int wire) {
    const int bw = 3 - wire;
    const float c = __cosf(0.5f * th), sn = __sinf(0.5f * th);
#pragma unroll
    for (int m = 0; m < 16; ++m) {
        if (!((m >> bw) & 1)) {
            const int m1 = m | (1 << bw);
            const float a = s[m], b = s[m1];
            s[m]  = c * a - sn * b;
            s[m1] = sn * a + c * b;
        }
    }
}

// U is stored row-major in LDS with stride 17 (17 coprime to 64 banks).
#define U_STRIDE 17

__global__ __launch_bounds__(256) void quanv_wmma_kernel(
        const float* __restrict__ x,       // (B,1,28,28) f32
        const float* __restrict__ params,  // (2,2) f32
        float* __restrict__ out,           // (B, 784) f32
        int total_patches) {
    __shared__ float Uld[16 * U_STRIDE];

    const int tid = threadIdx.x;

    // --- build fixed 16x16 circuit matrix U: thread j computes column j ---
    if (tid < 16) {
        float s[16];
#pragma unroll
        for (int m = 0; m < 16; ++m) s[m] = (m == tid) ? 1.0f : 0.0f;
        const float p00 = params[0], p01 = params[1];
        const float p10 = params[2], p11 = params[3];
        // layer 0: CNOT(0,1) RY(p00,w0) CNOT(0,1) RY(p01,w0)
        apply_cnot16(s, 0, 1);
        apply_ry16(s, p00, 0);
        apply_cnot16(s, 0, 1);
        apply_ry16(s, p01, 0);
        // layer 1: CNOT(1,2) RY(p10,w1) CNOT(1,2) RY(p11,w1)
        apply_cnot16(s, 1, 2);
        apply_ry16(s, p10, 1);
        apply_cnot16(s, 1, 2);
        apply_ry16(s, p11, 1);
#pragma unroll
        for (int m = 0; m < 16; ++m) Uld[m * U_STRIDE + tid] = s[m];
    }
    __syncthreads();

    // --- per-wave: 16 patches, W = U * V via 4x V_WMMA_F32_16X16X4_F32 ---
    const int lane = tid & 31;
    const bool hi  = (lane & 16) != 0;   // high half-wave: q2 = 1 states
    const int sub  = lane & 15;          // patch slot within tile / U row
    const int wave = tid >> 5;

    int patch = (blockIdx.x * 8 + wave) * 16 + sub;
    if (patch >= total_patches) patch = total_patches - 1;  // keep EXEC full

    const int img = patch / 196;
    const int pp  = patch - img * 196;
    const int pi  = pp / 14;
    const int pj  = pp - pi * 14;

    // patch pixels = RY angles: a(r,c) = x[img][2*pi+r][2*pj+c]
    const float* bp = x + ((long)img * 28 + 2 * pi) * 28 + 2 * pj;
    const float a0 = bp[0],  a1 = bp[1];
    const float a2 = bp[28], a3 = bp[29];

    const float c0 = __cosf(0.5f * a0), s0 = __sinf(0.5f * a0);
    const float c1 = __cosf(0.5f * a1), s1 = __sinf(0.5f * a1);
    const float c2 = __cosf(0.5f * a2), s2 = __sinf(0.5f * a2);
    const float c3 = __cosf(0.5f * a3), s3 = __sinf(0.5f * a3);

    // product-state vector v[m] = t0[q0]*t1[q1]*t2[q2]*t3[q3]
    // B-fragment chunk k = 2*q0+q1 ; lane half selects q2 ; {x,y} = q3={0,1}
    const float p2 = hi ? s2 : c2;
    const float pre0 = c0 * c1 * p2;
    const float pre1 = c0 * s1 * p2;
    const float pre2 = s0 * c1 * p2;
    const float pre3 = s0 * s1 * p2;
    v2f B0 = { pre0 * c3, pre0 * s3 };
    v2f B1 = { pre1 * c3, pre1 * s3 };
    v2f B2 = { pre2 * c3, pre2 * s3 };
    v2f B3 = { pre3 * c3, pre3 * s3 };

    // A-fragment: lane holds U[row][4k+off], U[row][4k+off+1], off = hi?2:0
    const int row = sub;
    const int off = hi ? 2 : 0;
    const float* ur = &Uld[row * U_STRIDE + off];
    v2f A0 = { ur[0],  ur[1]  };
    v2f A1 = { ur[4],  ur[5]  };
    v2f A2 = { ur[8],  ur[9]  };
    v2f A3 = { ur[12], ur[13] };

    v8f acc = {};
    acc = __builtin_amdgcn_wmma_f32_16x16x4_f32(false, A0, false, B0, (short)0, acc, false, false);
    acc = __builtin_amdgcn_wmma_f32_16x16x4_f32(false, A1, false, B1, (short)0, acc, false, false);
    acc = __builtin_amdgcn_wmma_f32_16x16x4_f32(false, A2, false, B2, (short)0, acc, false, false);
    acc = __builtin_amdgcn_wmma_f32_16x16x4_f32(false, A3, false, B3, (short)0, acc, false, false);

    // D layout: vgpr j -> output state m = j (+8 on high lanes), N = sub.
    // z_i = sum_m sign(bit_{3-i}(m)) * W[m]^2
    float sum0 = 0.f, z1p = 0.f, z2p = 0.f, z3p = 0.f;
#pragma unroll
    for (int v = 0; v < 8; ++v) {
        const float w2 = acc[v] * acc[v];
        sum0 += w2;
        z1p += ((v >> 2) & 1) ? -w2 : w2;  // q1 = bit2
        z2p += ((v >> 1) & 1) ? -w2 : w2;  // q2 = bit1
        z3p += (v & 1)        ? -w2 : w2;  // q3 = bit0
    }
    const float z0p = hi ? -sum0 : sum0;   // q0 = bit3 (lane half)

    const float z0 = z0p + __shfl_xor(z0p, 16, 32);
    const float z1 = z1p + __shfl_xor(z1p, 16, 32);
    const float z2 = z2p + __shfl_xor(z2p, 16, 32);
    const float z3 = z3p + __shfl_xor(z3p, 16, 32);

    if (!hi) {
        // out[img][pp*4 + i] == out flat at patch*4 + i
        float4 o = make_float4(z0, z1, z2, z3);
        *reinterpret_cast<float4*>(out + (long)patch * 4) = o;
    }
}

extern "C" void kernel_launch(void* const* d_in, const int* in_sizes, int n_in,
                              void* d_out, int out_size, void* d_ws, size_t ws_size,
                              hipStream_t stream) {
    const float* x      = (const float*)d_in[0];   // (8192,1,28,28)
    const float* params = (const float*)d_in[1];   // (2,2)
    float* out          = (float*)d_out;           // (8192,784)

    const int total_patches = in_sizes[0] / 4;          // 196 per image * B
    const int patches_per_block = 16 * 8;               // 16 per wave, 8 waves
    const int blocks = (total_patches + patches_per_block - 1) / patches_per_block;

    hipLaunchKernelGGL(quanv_wmma_kernel, dim3(blocks), dim3(256), 0, stream,
                       x, params, out, total_patches);
}